// GALR_Encoder_52656299049112
// MI455X (gfx1250) — compile-verified
//
#include <hip/hip_runtime.h>

#define D 64
#define EPS 0.2f
#define TILE 256

__device__ __forceinline__ float sgnf(float x) {
    return (x > 0.0f) ? 1.0f : ((x < 0.0f) ? -1.0f : 0.0f);
}

// ---------------------------------------------------------------------------
// spmm: out[rows[e]] += vals[e] * x[cols[e]]  (COO scatter-add)
// Defined FIRST so the disasm snippet shows this (hot) kernel.
// 256 edges per block, staged into LDS with gfx1250 async global->LDS copies
// (ASYNCcnt path). Processing: one HALF-wave (16 lanes) per edge, float4 per
// lane -> one global_load_b128 per edge, four f32 atomic adds per lane.
// ---------------------------------------------------------------------------
__global__ __launch_bounds__(256) void spmm_kernel(
    const float* __restrict__ x, const int* __restrict__ rows,
    const int* __restrict__ cols, const float* __restrict__ vals,
    float* __restrict__ out, int E)
{
    __shared__ int   sr[TILE];
    __shared__ int   sc[TILE];
    __shared__ float sv[TILE];

    const int tid  = threadIdx.x;
    const int base = blockIdx.x * TILE;
    int n = E - base; if (n > TILE) n = TILE;

    if (tid < n) {
        // low 32 bits of the generic shared address == wave-relative LDS offset
        unsigned lr = (unsigned)(size_t)&sr[tid];
        unsigned lc = (unsigned)(size_t)&sc[tid];
        unsigned lv = (unsigned)(size_t)&sv[tid];
        const int*   pr = rows + base + tid;
        const int*   pc = cols + base + tid;
        const float* pv = vals + base + tid;
        asm volatile("global_load_async_to_lds_b32 %0, %1, off" :: "v"(lr), "v"(pr) : "memory");
        asm volatile("global_load_async_to_lds_b32 %0, %1, off" :: "v"(lc), "v"(pc) : "memory");
        asm volatile("global_load_async_to_lds_b32 %0, %1, off" :: "v"(lv), "v"(pv) : "memory");
    }
    asm volatile("s_wait_asynccnt 0" ::: "memory");
    __syncthreads();

    const int lane = tid & 31;
    const int sub  = lane >> 4;         // half-wave id: 0 or 1
    const int l16  = lane & 15;         // lane within half-wave
    const int e0   = (tid >> 5) * 32;   // wave w owns edges [32w, 32w+32)
    int e1 = e0 + 32; if (e1 > n) e1 = n;

    for (int e = e0 + sub; e < e1; e += 2) {
        const int   r = sr[e];
        const int   c = sc[e];
        const float v = sv[e];
        const float4 xv = *(const float4*)(x + (size_t)c * D + l16 * 4);
        float* o = out + (size_t)r * D + l16 * 4;
        // relaxed agent-scope f32 adds -> L2-side global_atomic_add_f32
        __hip_atomic_fetch_add(o + 0, v * xv.x, __ATOMIC_RELAXED, __HIP_MEMORY_SCOPE_AGENT);
        __hip_atomic_fetch_add(o + 1, v * xv.y, __ATOMIC_RELAXED, __HIP_MEMORY_SCOPE_AGENT);
        __hip_atomic_fetch_add(o + 2, v * xv.z, __ATOMIC_RELAXED, __HIP_MEMORY_SCOPE_AGENT);
        __hip_atomic_fetch_add(o + 3, v * xv.w, __ATOMIC_RELAXED, __HIP_MEMORY_SCOPE_AGENT);
    }
}

// ---------------------------------------------------------------------------
// post: per row (one wave32 per row):
//   e   = spmm result; tmp re-zeroed in place for next layer
//   n'  = noise / max(||noise||, 1e-12)       (wave32 shfl_xor reduction)
//   e  += sign(e) * n' * EPS ; ego = e ; acc += e (scaled 1/L on last layer)
//   cl  = e on the contrastive layer (k == LAYER_CL-1 == 0)
// ---------------------------------------------------------------------------
__global__ __launch_bounds__(256) void post_kernel(
    float* __restrict__ tmp, const float* __restrict__ nz,
    float* __restrict__ ego, float* __restrict__ acc, float* __restrict__ cl,
    int N, int write_cl, int is_last, float inv_layers)
{
    const int wave = (int)((blockIdx.x * 256u + threadIdx.x) >> 5);
    if (wave >= N) return;
    const int lane = threadIdx.x & 31;
    const size_t base = (size_t)wave * D + (size_t)lane * 2;

    float2 e  = *(float2*)(tmp + base);
    float2 nv = *(const float2*)(nz + base);
    *(float2*)(tmp + base) = make_float2(0.f, 0.f);   // re-arm accumulator

    float s = nv.x * nv.x + nv.y * nv.y;
    s += __shfl_xor(s, 16, 32);
    s += __shfl_xor(s, 8, 32);
    s += __shfl_xor(s, 4, 32);
    s += __shfl_xor(s, 2, 32);
    s += __shfl_xor(s, 1, 32);
    const float scale = EPS / fmaxf(sqrtf(s), 1e-12f);

    e.x += sgnf(e.x) * nv.x * scale;
    e.y += sgnf(e.y) * nv.y * scale;
    *(float2*)(ego + base) = e;

    float2 a = *(float2*)(acc + base);
    a.x += e.x; a.y += e.y;
    if (is_last) { a.x *= inv_layers; a.y *= inv_layers; }
    *(float2*)(acc + base) = a;

    if (write_cl) *(float2*)(cl + base) = e;
}

// ---------------------------------------------------------------------------
// init: ego = concat(user_emb, item_emb); acc = 0 (in d_out); tmp = 0
// ---------------------------------------------------------------------------
__global__ __launch_bounds__(256) void init_kernel(
    const float4* __restrict__ ue, const float4* __restrict__ ie,
    float4* __restrict__ ego, float4* __restrict__ acc, float4* __restrict__ tmp,
    int ud4, int nd4)
{
    int i = blockIdx.x * 256 + threadIdx.x;
    if (i >= nd4) return;
    float4 v = (i < ud4) ? ue[i] : ie[i - ud4];
    ego[i] = v;
    acc[i] = make_float4(0.f, 0.f, 0.f, 0.f);
    tmp[i] = make_float4(0.f, 0.f, 0.f, 0.f);
}

// ---------------------------------------------------------------------------
extern "C" void kernel_launch(void* const* d_in, const int* in_sizes, int n_in,
                              void* d_out, int out_size, void* d_ws, size_t ws_size,
                              hipStream_t stream)
{
    const float* ue    = (const float*)d_in[0];
    const float* ie    = (const float*)d_in[1];
    const int*   er    = (const int*)  d_in[2];
    const int*   ec    = (const int*)  d_in[3];
    const float* ev    = (const float*)d_in[4];
    const float* noise = (const float*)d_in[5];

    const int U = in_sizes[0] / D;
    const int I = in_sizes[1] / D;
    const int E = in_sizes[2];
    const int N = U + I;
    const size_t ND = (size_t)N * D;
    const int L = (int)((size_t)in_sizes[5] / ND);   // n_layers (3)

    float* ego = (float*)d_ws;          // [N, D]
    float* tmp = ego + ND;              // [N, D] spmm accumulator
    float* acc = (float*)d_out;         // rows [0, N): running sum -> final mean
    float* cl  = (float*)d_out + ND;    // rows [N, 2N): contrastive layer

    const int nd4 = (int)(ND / 4);
    const int ud4 = in_sizes[0] / 4;
    init_kernel<<<(nd4 + 255) / 256, 256, 0, stream>>>(
        (const float4*)ue, (const float4*)ie,
        (float4*)ego, (float4*)acc, (float4*)tmp, ud4, nd4);

    const int spmm_blocks = (E + TILE - 1) / TILE;
    const int post_blocks = (N + 7) / 8;            // 8 rows (waves) per block
    const float invL = 1.0f / (float)L;
    for (int k = 0; k < L; ++k) {
        spmm_kernel<<<spmm_blocks, 256, 0, stream>>>(ego, er, ec, ev, tmp, E);
        post_kernel<<<post_blocks, 256, 0, stream>>>(
            tmp, noise + (size_t)k * ND, ego, acc, cl,
            N, (k == 0) ? 1 : 0, (k == L - 1) ? 1 : 0, invL);
    }
}